// SelectiveSSM_89610197664095
// MI455X (gfx1250) — compile-verified
//
#include <hip/hip_runtime.h>
#include <hip/hip_bf16.h>
#include <math.h>

typedef float v2f __attribute__((ext_vector_type(2)));
typedef float v8f __attribute__((ext_vector_type(8)));
typedef unsigned int u32x4 __attribute__((ext_vector_type(4)));
typedef int i32x4 __attribute__((ext_vector_type(4)));
typedef int i32x8 __attribute__((ext_vector_type(8)));

#define D_MODEL 1024
#define D_STATE 16
#define DT_RANK 64
#define SEQ_L   4096
#define BATCH   2
#define ROWS    (BATCH * SEQ_L)   // 8192 token rows
#define CHUNK   128               // scan steps staged per TDM tile
#define NCHUNK  (SEQ_L / CHUNK)   // 32

__device__ __forceinline__ float siluf(float x)  { return x / (1.f + __expf(-x)); }
__device__ __forceinline__ float softplusf(float x) { return (x > 20.f) ? x : log1pf(__expf(x)); }
__device__ __forceinline__ float clampf(float x, float lo, float hi) { return fminf(fmaxf(x, lo), hi); }

// ---------------------------------------------------------------------------
// TDM helper: issue a tensor_load_to_lds of a 2-D fp32 tile
//   global tensor: rows x 16 (row stride 16 elements), tile: CHUNK rows x 16
//   D# packing per CDNA5 ISA
// ---------------------------------------------------------------------------
__device__ __forceinline__ void tdm_load_rows16(const float* gsrc, unsigned lds_off,
                                                unsigned rows_remaining) {
    unsigned long long ga = (unsigned long long)(uintptr_t)gsrc;
    u32x4 g0;
    g0.x = 1u;                                            // count=1, user mode, no gather
    g0.y = lds_off;                                       // LDS byte address
    g0.z = (unsigned)(ga & 0xFFFFFFFFu);                  // global_addr[31:0]
    g0.w = (unsigned)((ga >> 32) & 0x01FFFFFFu) | (2u << 30); // addr[56:32] | type=2

    i32x8 g1;
    g1[0] = (int)(2u << 16);                              // wg_mask=0, data_size=4B
    g1[1] = (int)((D_STATE & 0xFFFFu) << 16);             // tensor_dim0 = 16 (lo bits @48)
    g1[2] = (int)((rows_remaining & 0xFFFFu) << 16);      // dim0 hi=0 | tensor_dim1 lo @80
    g1[3] = (int)(((rows_remaining >> 16) & 0xFFFFu) | (D_STATE << 16)); // dim1 hi | tile_dim0
    g1[4] = (int)(CHUNK & 0xFFFFu);                       // tile_dim1 = CHUNK, tile_dim2 = 0
    g1[5] = (int)D_STATE;                                 // tensor_dim0_stride = 16 elements
    g1[6] = 0;                                            // stride hi / dim1_stride = 0
    g1[7] = 0;

    i32x4 gz = {0, 0, 0, 0};                              // groups 2/3: unused (2-D)
#if __clang_major__ >= 23
    i32x8 gz8 = {0, 0, 0, 0, 0, 0, 0, 0};
    __builtin_amdgcn_tensor_load_to_lds(g0, g1, gz, gz, gz8, 0);
#else
    __builtin_amdgcn_tensor_load_to_lds(g0, g1, gz, gz, 0);
#endif
}

// ---------------------------------------------------------------------------
// Kernel 1: depthwise causal conv (K=4, left pad 3) + bias + SiLU
// ---------------------------------------------------------------------------
__global__ void conv_silu_kernel(const float* __restrict__ x,
                                 const float* __restrict__ w,
                                 const float* __restrict__ bias,
                                 float* __restrict__ xc) {
    int idx = blockIdx.x * blockDim.x + threadIdx.x;      // over B*L*D
    if (idx >= ROWS * D_MODEL) return;
    int d  = idx & (D_MODEL - 1);
    int bl = idx >> 10;
    int l  = bl & (SEQ_L - 1);
    int b  = bl >> 12;
    const float* xcol = x + (size_t)b * SEQ_L * D_MODEL + d;
    float acc = bias[d];
#pragma unroll
    for (int k = 0; k < 4; ++k) {
        int ls = l + k - 3;                               // causal: left pad 3
        float xv = (ls >= 0) ? xcol[(size_t)ls * D_MODEL] : 0.f;
        acc = fmaf(w[d * 4 + k], xv, acc);
    }
    xc[idx] = siluf(acc);
}

// ---------------------------------------------------------------------------
// Kernel 2: fused projections via fp32 WMMA (V_WMMA_F32_16X16X4_F32).
// ---------------------------------------------------------------------------
__global__ void proj_wmma_kernel(const float* __restrict__ xc,
                                 const float* __restrict__ Wdt1,  // (64,1024)
                                 const float* __restrict__ WB,    // (16,1024)
                                 const float* __restrict__ WC,    // (16,1024)
                                 float* __restrict__ dt_raw,      // (8192,64)
                                 float* __restrict__ Bssm,        // (8192,16)
                                 float* __restrict__ Cssm) {      // (8192,16)
    const int lane = threadIdx.x & 31;
    const int wave = threadIdx.x >> 5;        // 0..5
    const int row0 = blockIdx.x * 16;
    const int m    = lane & 15;
    const int hi   = lane >> 4;
    const int koff = hi * 2;

    const float* Wsel;
    int nbase;
    if (wave < 4)      { Wsel = Wdt1; nbase = wave * 16; }
    else if (wave == 4){ Wsel = WB;   nbase = 0; }
    else               { Wsel = WC;   nbase = 0; }
    const int n = nbase + m;

    const float* arow = xc   + (size_t)(row0 + m) * D_MODEL;
    const float* brow = Wsel + (size_t)n * D_MODEL;

    v8f acc = {};
    for (int k0 = 0; k0 < D_MODEL; k0 += 4) {
        v2f a  = *(const v2f*)(arow + k0 + koff);
        v2f bf = *(const v2f*)(brow + k0 + koff);
        acc = __builtin_amdgcn_wmma_f32_16x16x4_f32(false, a, false, bf,
                                                    (short)0, acc, false, false);
    }

    float* out; int stride;
    if (wave < 4)      { out = dt_raw; stride = DT_RANK; }
    else if (wave == 4){ out = Bssm;   stride = D_STATE; }
    else               { out = Cssm;   stride = D_STATE; }
#pragma unroll
    for (int v = 0; v < 8; ++v) {
        int r = row0 + v + hi * 8;
        out[(size_t)r * stride + n] = acc[v];
    }
}

// ---------------------------------------------------------------------------
// Kernel 3: dt = softplus(dt_raw (8192x64) x W_dt2^T (64x1024) + b_dt)
// ---------------------------------------------------------------------------
__global__ void dtproj_wmma_kernel(const float* __restrict__ dt_raw, // (8192,64)
                                   const float* __restrict__ Wdt2,   // (1024,64)
                                   const float* __restrict__ b_dt,   // (1024)
                                   float* __restrict__ dt) {         // (8192,1024)
    const int lane = threadIdx.x & 31;
    const int wave = threadIdx.x >> 5;            // 0..3
    const int gw   = blockIdx.x * 4 + wave;       // 0..8191
    const int rowtile = gw >> 4;                  // 0..511
    const int cg      = gw & 15;                  // 64-col group
    const int row0 = rowtile * 16;
    const int m    = lane & 15;
    const int hi   = lane >> 4;
    const int koff = hi * 2;

    const float* arow = dt_raw + (size_t)(row0 + m) * DT_RANK;
    v8f acc[4] = {{}, {}, {}, {}};
#pragma unroll 4
    for (int k0 = 0; k0 < DT_RANK; k0 += 4) {
        v2f a = *(const v2f*)(arow + k0 + koff);
#pragma unroll
        for (int t = 0; t < 4; ++t) {
            int n = cg * 64 + t * 16 + m;
            v2f bf = *(const v2f*)(Wdt2 + (size_t)n * DT_RANK + k0 + koff);
            acc[t] = __builtin_amdgcn_wmma_f32_16x16x4_f32(false, a, false, bf,
                                                           (short)0, acc[t], false, false);
        }
    }
#pragma unroll
    for (int t = 0; t < 4; ++t) {
        int n = cg * 64 + t * 16 + m;
        float bb = b_dt[n];
#pragma unroll
        for (int v = 0; v < 8; ++v) {
            int r = row0 + v + hi * 8;
            dt[(size_t)r * D_MODEL + n] = softplusf(acc[t][v] + bb);
        }
    }
}

// ---------------------------------------------------------------------------
// Kernel 4: selective scan, TDM double-buffered.
// One lane per (b,d,n); block = 256 threads = 16 d-values of one batch b.
// TDM streams CHUNK-step tiles of B_ssm/C_ssm into LDS (TENSORcnt),
// compute on chunk c overlaps DMA of chunk c+1. y-dot via 16-lane shfl tree.
// ---------------------------------------------------------------------------
__global__ void scan_kernel(const float* __restrict__ xc,
                            const float* __restrict__ dt,
                            const float* __restrict__ Bssm,
                            const float* __restrict__ Cssm,
                            const float* __restrict__ A_log,
                            const float* __restrict__ Dp,
                            float* __restrict__ y) {
    __shared__ __attribute__((aligned(128))) float Bbuf[2][CHUNK][D_STATE];
    __shared__ __attribute__((aligned(128))) float Cbuf[2][CHUNK][D_STATE];

    int tid = blockIdx.x * blockDim.x + threadIdx.x;   // 0..32767
    int n = tid & 15;
    int d = (tid >> 4) & (D_MODEL - 1);
    int b = tid >> 14;
    size_t base = (size_t)b * SEQ_L;

    float A  = -__expf(A_log[d * D_STATE + n]);
    float Dd = Dp[d];
    float h  = 0.f;

    unsigned ldsB0 = (unsigned)(uintptr_t)&Bbuf[0][0][0];
    unsigned ldsB1 = (unsigned)(uintptr_t)&Bbuf[1][0][0];
    unsigned ldsC0 = (unsigned)(uintptr_t)&Cbuf[0][0][0];
    unsigned ldsC1 = (unsigned)(uintptr_t)&Cbuf[1][0][0];

    // prime the pipeline: chunk 0 -> buffer 0
    if (threadIdx.x == 0) {
        tdm_load_rows16(Bssm + base * D_STATE, ldsB0, (unsigned)(ROWS - base));
        tdm_load_rows16(Cssm + base * D_STATE, ldsC0, (unsigned)(ROWS - base));
    }

    for (int c = 0; c < NCHUNK; ++c) {
        int bi = c & 1;
        if (threadIdx.x == 0) {
            if (c + 1 < NCHUNK) {
                size_t r1 = base + (size_t)(c + 1) * CHUNK;
                tdm_load_rows16(Bssm + r1 * D_STATE, bi ? ldsB0 : ldsB1,
                                (unsigned)(ROWS - r1));
                tdm_load_rows16(Cssm + r1 * D_STATE, bi ? ldsC0 : ldsC1,
                                (unsigned)(ROWS - r1));
                __builtin_amdgcn_s_wait_tensorcnt(2);   // chunk c landed; c+1 in flight
            } else {
                __builtin_amdgcn_s_wait_tensorcnt(0);
            }
        }
        __syncthreads();                                // release readers

        size_t rowc = base + (size_t)c * CHUNK;
        for (int s = 0; s < CHUNK; ++s) {
            size_t rd = rowc + s;
            float x_t  = xc[rd * D_MODEL + d];
            float dt_t = dt[rd * D_MODEL + d];
            float B_t  = Bbuf[bi][s][n];
            float C_t  = Cbuf[bi][s][n];

            float dtA = clampf(dt_t * A, -10.f, 0.f);
            float Ab  = fminf(__expf(dtA), 0.999f);     // exp(dtA) >= 0 already
            float Bb  = clampf(clampf(dt_t, 0.f, 1.f) * B_t, -10.f, 10.f);
            h = clampf(fmaf(Ab, h, Bb * x_t), -100.f, 100.f);

            float p = h * C_t;
            p += __shfl_xor(p, 1, 16);
            p += __shfl_xor(p, 2, 16);
            p += __shfl_xor(p, 4, 16);
            p += __shfl_xor(p, 8, 16);
            if (n == 0)
                y[rd * D_MODEL + d] = clampf(p + Dd * x_t, -50.f, 50.f);
        }
        __syncthreads();        // all readers done before buffer bi is re-filled
    }
}

// ---------------------------------------------------------------------------
// Launch
// ---------------------------------------------------------------------------
extern "C" void kernel_launch(void* const* d_in, const int* in_sizes, int n_in,
                              void* d_out, int out_size, void* d_ws, size_t ws_size,
                              hipStream_t stream) {
    const float* x      = (const float*)d_in[0];  // (B,L,D)
    const float* A_log  = (const float*)d_in[1];  // (D,N)
    const float* Dp     = (const float*)d_in[2];  // (D)
    const float* W_dt1  = (const float*)d_in[3];  // (R,D)
    const float* W_dt2  = (const float*)d_in[4];  // (D,R)
    const float* b_dt   = (const float*)d_in[5];  // (D)
    const float* W_B    = (const float*)d_in[6];  // (N,D)
    const float* W_C    = (const float*)d_in[7];  // (N,D)
    const float* conv_w = (const float*)d_in[8];  // (D,1,4)
    const float* conv_b = (const float*)d_in[9];  // (D)
    float* y = (float*)d_out;                     // (B,L,D)

    // workspace layout (floats)
    float* ws     = (float*)d_ws;
    float* xc     = ws;                                   // 8192*1024
    float* dt_raw = xc + (size_t)ROWS * D_MODEL;          // 8192*64
    float* dtb    = dt_raw + (size_t)ROWS * DT_RANK;      // 8192*1024
    float* Bs     = dtb + (size_t)ROWS * D_MODEL;         // 8192*16
    float* Cs     = Bs + (size_t)ROWS * D_STATE;          // 8192*16

    // 1) conv + silu
    {
        int total = ROWS * D_MODEL;
        conv_silu_kernel<<<(total + 255) / 256, 256, 0, stream>>>(x, conv_w, conv_b, xc);
    }
    // 2) fused dt_raw / B_ssm / C_ssm projections (WMMA fp32)
    proj_wmma_kernel<<<ROWS / 16, 192, 0, stream>>>(xc, W_dt1, W_B, W_C, dt_raw, Bs, Cs);
    // 3) dt = softplus(dt_raw @ W_dt2^T + b_dt) (WMMA fp32)
    dtproj_wmma_kernel<<<(ROWS / 16) * (D_MODEL / 64) / 4, 128, 0, stream>>>(dt_raw, W_dt2, b_dt, dtb);
    // 4) selective scan with TDM double-buffering
    scan_kernel<<<(BATCH * D_MODEL * D_STATE) / 256, 256, 0, stream>>>(xc, dtb, Bs, Cs, A_log, Dp, y);
}